// GraphNN_38268158607515
// MI455X (gfx1250) — compile-verified
//
#include <hip/hip_runtime.h>

typedef __attribute__((ext_vector_type(2))) float v2f;
typedef __attribute__((ext_vector_type(8))) float v8f;

#define BATCH   16384
#define WIDTH   128
#define CHUNK   ((size_t)BATCH * WIDTH)   // floats per node activation
#define ROWS_PB 64                        // rows per block

// One level of the graph NN:
//   out = sum_{u=0..p-1} relu( A_u @ W_u + b_u )
// A_0 = x, A_u (u>=1) = ws + (u-1)*CHUNK.
//
// Block = 256 threads = 8 waves. Wave w handles rows [blk*64 + (w>>1)*16, +16)
// and column half (w&1): column tiles ctBase..ctBase+3 (64 cols).
//
// W is staged in LDS in WMMA-B fragment order so each fragment fetch is one
// conflict-free ds_load_b64 across a contiguous 256B block:
//   ldsw[(ks*8 + ct)*32 + lane] = { W[4ks + 2*(lane>>4) + 0][ct*16 + lane&15],
//                                   W[4ks + 2*(lane>>4) + 1][ct*16 + lane&15] }
// matching the 32-bit 4x16 B layout (VGPR0: K=k0 / K=k0+2, VGPR1: K=k0+1 / K=k0+3).
__global__ __launch_bounds__(256)
void graphnn_level_kernel(const float* __restrict__ x,
                          const float* __restrict__ ws,
                          const float* __restrict__ Wl,
                          const float* __restrict__ bl,
                          float* __restrict__ out,
                          int p)
{
    __shared__ v2f ldsw[32 * 8 * 32];     // 8192 * 8B = 64 KB

    const int tid    = threadIdx.x;
    const int lane   = tid & 31;
    const int wave   = tid >> 5;
    const int laneLo = lane & 15;
    const int laneHi = lane >> 4;         // 0 for lanes 0-15, 1 for 16-31
    const int rowBase = blockIdx.x * ROWS_PB + (wave >> 1) * 16;
    const int ctBase  = (wave & 1) * 4;   // column tiles ctBase..ctBase+3

    const v8f vzero = {0.f, 0.f, 0.f, 0.f, 0.f, 0.f, 0.f, 0.f};

    v8f acc[4];
#pragma unroll
    for (int ct = 0; ct < 4; ++ct) acc[ct] = vzero;

    for (int u = 0; u < p; ++u) {
        const float* A  = (u == 0) ? x : (ws + (size_t)(u - 1) * CHUNK);
        const float* We = Wl + (size_t)u * (WIDTH * WIDTH);
        const float* be = bl + (size_t)u * WIDTH;

        __syncthreads();   // previous W fully consumed
        // Cooperative fill of W into LDS in fragment order (32 entries/thread).
        for (int f = 0; f < 32; ++f) {
            int e   = f * 256 + tid;          // 0..8191
            int ks  = e >> 8;                 // k-step 0..31
            int rem = e & 255;
            int ct  = rem >> 5;               // column tile 0..7
            int ln  = rem & 31;               // lane slot
            int k   = 4 * ks + 2 * (ln >> 4);
            int n   = ct * 16 + (ln & 15);
            v2f w;
            w.x = We[(size_t)k * WIDTH + n];
            w.y = We[(size_t)(k + 1) * WIDTH + n];
            ldsw[e] = w;
        }
        __syncthreads();

        // Preload this wave's 16x128 A tile as 32 WMMA A-fragments.
        // 32-bit 16x4 A layout: lanes 0-15 hold K=k0,k0+1; lanes 16-31 hold K=k0+2,k0+3.
        v2f afrag[32];
        {
            const float* arow = A + (size_t)(rowBase + laneLo) * WIDTH + 2 * laneHi;
#pragma unroll
            for (int ks = 0; ks < 32; ++ks)
                afrag[ks] = *(const v2f*)(arow + 4 * ks);
        }

#pragma unroll
        for (int ct = 0; ct < 4; ++ct) {
            const int ctIdx = ctBase + ct;
            v8f t = vzero;
#pragma unroll
            for (int ks = 0; ks < 32; ++ks) {
                v2f bfrag = ldsw[(ks * 8 + ctIdx) * 32 + lane];
                t = __builtin_amdgcn_wmma_f32_16x16x4_f32(
                        /*neg_a=*/false, afrag[ks],
                        /*neg_b=*/false, bfrag,
                        /*c_mod=*/(short)0, t,
                        /*reuse_a=*/false, /*reuse_b=*/false);
            }
            float bias = be[ctIdx * 16 + laneLo];
#pragma unroll
            for (int i = 0; i < 8; ++i) {
                float h = t[i] + bias;
                acc[ct][i] += (h > 0.f) ? h : 0.f;
            }
        }
    }

    // C/D layout: VGPR i -> row (i + 8*laneHi), col = laneLo within the tile.
#pragma unroll
    for (int ct = 0; ct < 4; ++ct) {
        int col = (ctBase + ct) * 16 + laneLo;
#pragma unroll
        for (int i = 0; i < 8; ++i) {
            int row = rowBase + laneHi * 8 + i;
            out[(size_t)row * WIDTH + col] = acc[ct][i];
        }
    }
}

extern "C" void kernel_launch(void* const* d_in, const int* in_sizes, int n_in,
                              void* d_out, int out_size, void* d_ws, size_t ws_size,
                              hipStream_t stream)
{
    const float* x = (const float*)d_in[0];   // (16384, 128)
    const float* W = (const float*)d_in[1];   // (66, 128, 128)
    const float* b = (const float*)d_in[2];   // (66, 128)
    float* out = (float*)d_out;
    float* ws  = (float*)d_ws;                // activations for nodes 1..10 (80 MB)

    dim3 block(256);
    dim3 grid(BATCH / ROWS_PB);               // 256 blocks

    int start = 0;
    for (int v = 2; v <= 12; ++v) {
        int p = v - 1;
        const float* Wl = W + (size_t)start * WIDTH * WIDTH;
        const float* bl = b + (size_t)start * WIDTH;
        float* o = (v == 12) ? out : (ws + (size_t)(v - 2) * CHUNK);
        graphnn_level_kernel<<<grid, block, 0, stream>>>(x, ws, Wl, bl, o, p);
        start += p;
    }
}